// ERIC_38079180046809
// MI455X (gfx1250) — compile-verified
//
#include <hip/hip_runtime.h>
#include <hip/hip_bf16.h>
#include <math.h>

typedef __attribute__((ext_vector_type(2))) float v2f;
typedef __attribute__((ext_vector_type(8))) float v8f;

#define NNODES 65536
#define NEDGES 1048576
#define NB     512
#define DIN    32
#define FDIM   64

// ---------------------------------------------------------------------------
// CDNA5 fp32 WMMA: D[16,16] = A[16,4] * B[4,16] + C
// A frag: lane m = lane&15, kpair = (lane>>4)*2 -> a[i] = A[m][kpair+i]
// B frag: lane n = lane&15, kpair = (lane>>4)*2 -> b[i] = B[kpair+i][n]
// C/D:    reg r, lane: n = lane&15, m = r + 8*(lane>>4)
// ---------------------------------------------------------------------------
__device__ __forceinline__ v8f wmma4(v2f a, v2f b, v8f c) {
  return __builtin_amdgcn_wmma_f32_16x16x4_f32(false, a, false, b, (short)0, c,
                                               false, false);
}

__device__ __forceinline__ void atomic_add_f32(float* p, float v) {
  __hip_atomic_fetch_add(p, v, __ATOMIC_RELAXED, __HIP_MEMORY_SCOPE_AGENT);
}

#define VZERO8 {0.f,0.f,0.f,0.f,0.f,0.f,0.f,0.f}

// ---------------------------------------------------------------------------
__global__ void zero_kernel(float* __restrict__ p, int n) {
  int i = blockIdx.x * blockDim.x + threadIdx.x;
  if (i < n) p[i] = 0.0f;
}

// agg[dst, 0:din] += x[src, 0:din]; agg row stride fixed at 64
__global__ void scatter_kernel(const float* __restrict__ x,
                               const int* __restrict__ ei,
                               float* agg, int nE, int din, int xStride) {
  long tid = (long)blockIdx.x * blockDim.x + threadIdx.x;
  int chunks = din >> 2;
  long total = (long)nE * chunks;
  if (tid >= total) return;
  int e = (int)(tid / chunks);
  int c = (int)(tid % chunks);
  int s = ei[e];
  int d = ei[nE + e];
  const float4 v = *(const float4*)(x + (size_t)s * xStride + c * 4);
  float* ar = agg + (size_t)d * 64 + c * 4;
  atomic_add_f32(ar + 0, v.x);
  atomic_add_f32(ar + 1, v.y);
  atomic_add_f32(ar + 2, v.z);
  atomic_add_f32(ar + 3, v.w);
}

// ---------------------------------------------------------------------------
// Fused GIN layer: x_out = relu(BN(relu(((1+eps)x + agg) W1 + b1) W2 + b2))
// plus inner = relu(x_out Wi + bi) pooled (segment-sum over 128-node graphs).
// 128 threads = 4 waves; wave w owns node rows [blk*64 + w*16, +16).
// NOTE: xout may alias agg (in-place, per-row disjoint across blocks) -> no
// __restrict__ on those two.
// ---------------------------------------------------------------------------
__global__ __launch_bounds__(128) void gin_fused_kernel(
    const float* __restrict__ x, const float* agg, float* xout,
    float* pool,
    const float* __restrict__ W1, const float* __restrict__ b1,
    const float* __restrict__ W2, const float* __restrict__ b2,
    const float* __restrict__ gamma, const float* __restrict__ bnbeta,
    const float* __restrict__ epsP,
    const float* __restrict__ Wi, const float* __restrict__ bi,
    int din, int xStride) {
  __shared__ float sh[4][16][64];
  const int lane = threadIdx.x & 31;
  const int wv   = threadIdx.x >> 5;
  const int n15  = lane & 15;
  const int hf   = lane >> 4;
  const int row0 = blockIdx.x * 64 + wv * 16;
  const float ope = 1.0f + epsP[0];

  const float* xr = x   + (size_t)(row0 + n15) * xStride;
  const float* ar = agg + (size_t)(row0 + n15) * 64;

  const v8f vz = VZERO8;
  v8f acc[4] = {vz, vz, vz, vz};
  for (int k0 = 0; k0 < din; k0 += 4) {
    const int ka = k0 + hf * 2;
    v2f a;
    a.x = ope * xr[ka]     + ar[ka];
    a.y = ope * xr[ka + 1] + ar[ka + 1];
#pragma unroll
    for (int t = 0; t < 4; ++t) {
      const int col = t * 16 + n15;
      v2f b;
      b.x = W1[ka * FDIM + col];
      b.y = W1[(ka + 1) * FDIM + col];
      acc[t] = wmma4(a, b, acc[t]);
    }
  }
  // z = relu(. + b1) -> LDS (row-major 16x64 per wave)
#pragma unroll
  for (int t = 0; t < 4; ++t) {
    const int col = t * 16 + n15;
    const float bb = b1[col];
#pragma unroll
    for (int r = 0; r < 8; ++r)
      sh[wv][r + hf * 8][col] = fmaxf(acc[t][r] + bb, 0.0f);
  }
  __syncthreads();

  v8f acc2[4] = {vz, vz, vz, vz};
#pragma unroll
  for (int k0 = 0; k0 < FDIM; k0 += 4) {
    const int ka = k0 + hf * 2;
    v2f a;
    a.x = sh[wv][n15][ka];
    a.y = sh[wv][n15][ka + 1];
#pragma unroll
    for (int t = 0; t < 4; ++t) {
      const int col = t * 16 + n15;
      v2f b;
      b.x = W2[ka * FDIM + col];
      b.y = W2[(ka + 1) * FDIM + col];
      acc2[t] = wmma4(a, b, acc2[t]);
    }
  }
  __syncthreads();

  // BN (eval: mean0/var1) + relu; write to global and restage in LDS
  const float invs = rsqrtf(1.0f + 1e-5f);
#pragma unroll
  for (int t = 0; t < 4; ++t) {
    const int col = t * 16 + n15;
    const float bb = b2[col];
    const float gs = gamma[col] * invs;
    const float bt = bnbeta[col];
#pragma unroll
    for (int r = 0; r < 8; ++r) {
      const int mm = r + hf * 8;
      float v = fmaf(acc2[t][r] + bb, gs, bt);
      v = fmaxf(v, 0.0f);
      xout[(size_t)(row0 + mm) * 64 + col] = v;
      sh[wv][mm][col] = v;
    }
  }
  __syncthreads();

  // inner GEMM + relu + row-reduce + pooled atomic add
  v8f acc3[4] = {vz, vz, vz, vz};
#pragma unroll
  for (int k0 = 0; k0 < FDIM; k0 += 4) {
    const int ka = k0 + hf * 2;
    v2f a;
    a.x = sh[wv][n15][ka];
    a.y = sh[wv][n15][ka + 1];
#pragma unroll
    for (int t = 0; t < 4; ++t) {
      const int col = t * 16 + n15;
      v2f b;
      b.x = Wi[ka * FDIM + col];
      b.y = Wi[(ka + 1) * FDIM + col];
      acc3[t] = wmma4(a, b, acc3[t]);
    }
  }
  const int gid = row0 >> 7;  // 128 nodes per graph, tiles aligned
#pragma unroll
  for (int t = 0; t < 4; ++t) {
    const int col = t * 16 + n15;
    const float bb = bi[col];
    float s = 0.0f;
#pragma unroll
    for (int r = 0; r < 8; ++r) s += fmaxf(acc3[t][r] + bb, 0.0f);
    s += __shfl_xor(s, 16, 32);
    if (hf == 0) atomic_add_f32(&pool[gid * FDIM + col], s);
  }
}

// ---------------------------------------------------------------------------
// o = relu(p Wo + bo) for both sides; diff = exp(-(o1-o2)^2) into diff_rep col
// block. Rows are graphs (B=512).
// ---------------------------------------------------------------------------
__global__ __launch_bounds__(128) void outer_diff_kernel(
    const float* __restrict__ p1, const float* __restrict__ p2,
    const float* __restrict__ Wo, const float* __restrict__ bo,
    float* __restrict__ o1, float* __restrict__ o2,
    float* __restrict__ diff, int layerOff) {
  const int lane = threadIdx.x & 31;
  const int wv   = threadIdx.x >> 5;
  const int n15  = lane & 15;
  const int hf   = lane >> 4;
  const int row0 = blockIdx.x * 64 + wv * 16;
  const v8f vz = VZERO8;
  v8f a1[4] = {vz, vz, vz, vz};
  v8f a2[4] = {vz, vz, vz, vz};
  const float* r1 = p1 + (size_t)(row0 + n15) * FDIM;
  const float* r2 = p2 + (size_t)(row0 + n15) * FDIM;
#pragma unroll
  for (int k0 = 0; k0 < FDIM; k0 += 4) {
    const int ka = k0 + hf * 2;
    v2f av1, av2;
    av1.x = r1[ka]; av1.y = r1[ka + 1];
    av2.x = r2[ka]; av2.y = r2[ka + 1];
#pragma unroll
    for (int t = 0; t < 4; ++t) {
      const int col = t * 16 + n15;
      v2f b;
      b.x = Wo[ka * FDIM + col];
      b.y = Wo[(ka + 1) * FDIM + col];
      a1[t] = wmma4(av1, b, a1[t]);
      a2[t] = wmma4(av2, b, a2[t]);
    }
  }
#pragma unroll
  for (int t = 0; t < 4; ++t) {
    const int col = t * 16 + n15;
    const float bb = bo[col];
#pragma unroll
    for (int r = 0; r < 8; ++r) {
      const int row = row0 + r + hf * 8;
      const float v1 = fmaxf(a1[t][r] + bb, 0.0f);
      const float v2 = fmaxf(a2[t][r] + bb, 0.0f);
      o1[row * FDIM + col] = v1;
      o2[row * FDIM + col] = v2;
      const float d = v1 - v2;
      diff[row * 192 + layerOff + col] = __expf(-d * d);
    }
  }
}

// ---------------------------------------------------------------------------
// score_rep = tanh(relu(diff[512,192] W1[192,96] + b1) W2[96,96] + b2)
// ---------------------------------------------------------------------------
__global__ __launch_bounds__(128) void conv_stack_kernel(
    const float* __restrict__ diff,
    const float* __restrict__ W1, const float* __restrict__ b1,
    const float* __restrict__ W2, const float* __restrict__ b2,
    float* __restrict__ srep) {
  __shared__ float sh[4][16][96];
  const int lane = threadIdx.x & 31;
  const int wv   = threadIdx.x >> 5;
  const int n15  = lane & 15;
  const int hf   = lane >> 4;
  const int row0 = blockIdx.x * 64 + wv * 16;
  const v8f vz = VZERO8;
  v8f acc[6] = {vz, vz, vz, vz, vz, vz};
  const float* dr = diff + (size_t)(row0 + n15) * 192;
  for (int k0 = 0; k0 < 192; k0 += 4) {
    const int ka = k0 + hf * 2;
    v2f a;
    a.x = dr[ka]; a.y = dr[ka + 1];
#pragma unroll
    for (int t = 0; t < 6; ++t) {
      const int col = t * 16 + n15;
      v2f b;
      b.x = W1[ka * 96 + col];
      b.y = W1[(ka + 1) * 96 + col];
      acc[t] = wmma4(a, b, acc[t]);
    }
  }
#pragma unroll
  for (int t = 0; t < 6; ++t) {
    const int col = t * 16 + n15;
    const float bb = b1[col];
#pragma unroll
    for (int r = 0; r < 8; ++r)
      sh[wv][r + hf * 8][col] = fmaxf(acc[t][r] + bb, 0.0f);
  }
  __syncthreads();
  v8f acc2[6] = {vz, vz, vz, vz, vz, vz};
  for (int k0 = 0; k0 < 96; k0 += 4) {
    const int ka = k0 + hf * 2;
    v2f a;
    a.x = sh[wv][n15][ka];
    a.y = sh[wv][n15][ka + 1];
#pragma unroll
    for (int t = 0; t < 6; ++t) {
      const int col = t * 16 + n15;
      v2f b;
      b.x = W2[ka * 96 + col];
      b.y = W2[(ka + 1) * 96 + col];
      acc2[t] = wmma4(a, b, acc2[t]);
    }
  }
#pragma unroll
  for (int t = 0; t < 6; ++t) {
    const int col = t * 16 + n15;
    const float bb = b2[col];
#pragma unroll
    for (int r = 0; r < 8; ++r)
      srep[(row0 + r + hf * 8) * 96 + col] = tanhf(acc2[t][r] + bb);
  }
}

// ---------------------------------------------------------------------------
// NTN: sim_rep[b,t] = relu( o1 W[:,:,t] o2^T + [o1,o2].Wb[t,:] + bias[t] )
// Tiny (67 MFLOP) -> scalar kernel, one thread per (b,t).
// ---------------------------------------------------------------------------
__global__ void ntn_kernel(const float* __restrict__ o1,
                           const float* __restrict__ o2,
                           const float* __restrict__ W,
                           const float* __restrict__ Wb,
                           const float* __restrict__ bias,
                           float* __restrict__ simrep) {
  const int tid = blockIdx.x * blockDim.x + threadIdx.x;  // 8192
  const int b = tid >> 4;
  const int t = tid & 15;
  const float* O1 = o1 + b * FDIM;
  const float* O2 = o2 + b * FDIM;
  float s = 0.0f;
  for (int i = 0; i < FDIM; ++i) {
    const float a = O1[i];
    const float* Wr = W + i * (FDIM * 16) + t;
    float ss = 0.0f;
    for (int j = 0; j < FDIM; ++j) ss += O2[j] * Wr[j * 16];
    s += a * ss;
  }
  const float* wb = Wb + t * 128;
  for (int k = 0; k < FDIM; ++k) s += O1[k] * wb[k] + O2[k] * wb[64 + k];
  simrep[b * 16 + t] = fmaxf(s + bias[t], 0.0f);
}

// ---------------------------------------------------------------------------
// Final scalar heads + blend. One thread per graph pair.
// ---------------------------------------------------------------------------
__global__ void heads_kernel(const float* __restrict__ simrep,
                             const float* __restrict__ srep,
                             const float* __restrict__ ssW1, const float* __restrict__ ssb1,
                             const float* __restrict__ ssW2, const float* __restrict__ ssb2,
                             const float* __restrict__ slW1, const float* __restrict__ slb1,
                             const float* __restrict__ slW2, const float* __restrict__ slb2,
                             const float* __restrict__ alpha,
                             const float* __restrict__ betap,
                             float* __restrict__ out) {
  const int b = blockIdx.x * blockDim.x + threadIdx.x;
  if (b >= NB) return;
  float h[16];
  // sim head: [16]->[16]->1
  for (int u = 0; u < 16; ++u) {
    float a = ssb1[u];
    for (int k = 0; k < 16; ++k) a += simrep[b * 16 + k] * ssW1[k * 16 + u];
    h[u] = fmaxf(a, 0.0f);
  }
  float s2 = ssb2[0];
  for (int u = 0; u < 16; ++u) s2 += h[u] * ssW2[u];
  const float sim_score = 1.0f / (1.0f + __expf(-s2));
  // score head: [96]->[16]->1
  for (int u = 0; u < 16; ++u) {
    float a = slb1[u];
    for (int k = 0; k < 96; ++k) a += srep[b * 96 + k] * slW1[k * 16 + u];
    h[u] = fmaxf(a, 0.0f);
  }
  float q = slb2[0];
  for (int u = 0; u < 16; ++u) q += h[u] * slW2[u];
  const float score = 1.0f / (1.0f + __expf(-q));
  out[b] = alpha[0] * score + betap[0] * sim_score;
}

// ---------------------------------------------------------------------------
extern "C" void kernel_launch(void* const* d_in, const int* in_sizes, int n_in,
                              void* d_out, int out_size, void* d_ws, size_t ws_size,
                              hipStream_t stream) {
  if (n_in < 56) return;
  auto F32 = [&](int i) { return (const float*)d_in[i]; };
  auto I32 = [&](int i) { return (const int*)d_in[i]; };

  const float *f1, *f2, *alpha, *betap;
  const int *e1, *e2;
  const float *csW1, *csW2, *csb1, *csb2;
  const float *ginW1[3], *ginW2[3], *ginb1[3], *ginb2[3];
  const float *ginBeta[3], *ginEps[3], *ginGamma[3];
  const float *innW[3], *innB[3], *outW[3], *outB[3];
  const float *ntnW, *ntnWb, *ntnBias;
  const float *scW1, *scW2, *scb1, *scb2;
  const float *ssW1, *ssW2, *ssb1, *ssb2;

  // Two candidate flatten orders; fingerprint via in_sizes[0]:
  //   jax tree_flatten (dict keys sorted): batch_1 (65536) first
  //   raw insertion order: features_1 (65536*32) first
  const bool sortedLayout = (in_sizes[0] == NNODES);
  if (sortedLayout) {
    e1 = I32(2); e2 = I32(3); f1 = F32(4); f2 = F32(5);
    alpha = F32(6); betap = F32(7);
    csW1 = F32(8); csW2 = F32(9); csb1 = F32(10); csb2 = F32(11);
    for (int i = 0; i < 3; ++i) {
      const int b = 12 + 7 * i;  // W1,W2,b1,b2,beta,eps,gamma (sorted)
      ginW1[i] = F32(b); ginW2[i] = F32(b + 1);
      ginb1[i] = F32(b + 2); ginb2[i] = F32(b + 3);
      ginBeta[i] = F32(b + 4); ginEps[i] = F32(b + 5); ginGamma[i] = F32(b + 6);
    }
    for (int i = 0; i < 3; ++i) { innW[i] = F32(33 + 2 * i); innB[i] = F32(34 + 2 * i); }
    ntnW = F32(39); ntnWb = F32(40); ntnBias = F32(41);
    for (int i = 0; i < 3; ++i) { outW[i] = F32(42 + 2 * i); outB[i] = F32(43 + 2 * i); }
    scW1 = F32(48); scW2 = F32(49); scb1 = F32(50); scb2 = F32(51);
    ssW1 = F32(52); ssW2 = F32(53); ssb1 = F32(54); ssb2 = F32(55);
  } else {
    f1 = F32(0); f2 = F32(1); e1 = I32(2); e2 = I32(3);
    for (int i = 0; i < 3; ++i) {
      const int b = 6 + 7 * i;  // eps,W1,b1,W2,b2,gamma,beta (insertion)
      ginEps[i] = F32(b); ginW1[i] = F32(b + 1); ginb1[i] = F32(b + 2);
      ginW2[i] = F32(b + 3); ginb2[i] = F32(b + 4);
      ginGamma[i] = F32(b + 5); ginBeta[i] = F32(b + 6);
    }
    for (int i = 0; i < 3; ++i) { innW[i] = F32(27 + 2 * i); innB[i] = F32(28 + 2 * i); }
    for (int i = 0; i < 3; ++i) { outW[i] = F32(33 + 2 * i); outB[i] = F32(34 + 2 * i); }
    ntnW = F32(39); ntnWb = F32(40); ntnBias = F32(41);
    csW1 = F32(42); csb1 = F32(43); csW2 = F32(44); csb2 = F32(45);
    scW1 = F32(46); scb1 = F32(47); scW2 = F32(48); scb2 = F32(49);
    ssW1 = F32(50); ssb1 = F32(51); ssW2 = F32(52); ssb2 = F32(53);
    alpha = F32(54); betap = F32(55);
  }

  // workspace carve
  float* ws = (float*)d_ws;
  size_t off = 0;
  auto carve = [&](size_t n) { float* p = ws + off; off += n; return p; };
  float* bufA[2]; float* bufB[2];
  bufA[0] = carve((size_t)NNODES * 64); bufB[0] = carve((size_t)NNODES * 64);
  bufA[1] = carve((size_t)NNODES * 64); bufB[1] = carve((size_t)NNODES * 64);
  float* p1 = carve(NB * 64); float* p2 = carve(NB * 64);
  float* o1 = carve(NB * 64); float* o2 = carve(NB * 64);
  float* diff   = carve(NB * 192);
  float* srep   = carve(NB * 96);
  float* simrep = carve(NB * 16);

  const float* xin[2] = {f1, f2};
  const int* eidx[2] = {e1, e2};
  float* pools[2] = {p1, p2};

  for (int l = 0; l < 3; ++l) {
    const int din = (l == 0) ? DIN : FDIM;
    const int xStride = din;
    zero_kernel<<<(NB * 64 + 255) / 256, 256, 0, stream>>>(p1, NB * 64);
    zero_kernel<<<(NB * 64 + 255) / 256, 256, 0, stream>>>(p2, NB * 64);
    for (int s = 0; s < 2; ++s) {
      const float* x; float* aggb; float* outb;
      if (l == 0)      { x = xin[s];  aggb = bufA[s]; outb = bufA[s]; }
      else if (l == 1) { x = bufA[s]; aggb = bufB[s]; outb = bufB[s]; }
      else             { x = bufB[s]; aggb = bufA[s]; outb = bufA[s]; }
      const int aggN = NNODES * 64;
      zero_kernel<<<(aggN + 255) / 256, 256, 0, stream>>>(aggb, aggN);
      const long total = (long)NEDGES * (din / 4);
      scatter_kernel<<<(int)((total + 255) / 256), 256, 0, stream>>>(
          x, eidx[s], aggb, NEDGES, din, xStride);
      gin_fused_kernel<<<NNODES / 64, 128, 0, stream>>>(
          x, aggb, outb, pools[s],
          ginW1[l], ginb1[l], ginW2[l], ginb2[l],
          ginGamma[l], ginBeta[l], ginEps[l],
          innW[l], innB[l], din, xStride);
    }
    outer_diff_kernel<<<NB / 64, 128, 0, stream>>>(p1, p2, outW[l], outB[l],
                                                   o1, o2, diff, l * 64);
  }
  conv_stack_kernel<<<NB / 64, 128, 0, stream>>>(diff, csW1, csb1, csW2, csb2, srep);
  ntn_kernel<<<(NB * 16) / 256, 256, 0, stream>>>(o1, o2, ntnW, ntnWb, ntnBias, simrep);
  heads_kernel<<<(NB + 255) / 256, 256, 0, stream>>>(
      simrep, srep, ssW1, ssb1, ssW2, ssb2, scW1, scb1, scW2, scb2,
      alpha, betap, (float*)d_out);
}